// FactoredMinigridImageFeatureExtractor_1743756722323
// MI455X (gfx1250) — compile-verified
//
#include <hip/hip_runtime.h>
#include <hip/hip_bf16.h>
#include <math.h>

// Problem constants (from reference): x is (2048, 6, 128, 96) f32,
// output is (2048*6, 768) f32 where 768 = 32*24.
#define IN_H   128
#define IN_W   96
#define OUT_H  32
#define OUT_W  24
#define IMG_ELEMS (IN_H * IN_W)     // 12288 floats per image
#define OUT_ELEMS (OUT_H * OUT_W)   // 768 floats per image
#define WAVES_PER_BLOCK 8

typedef float v2f __attribute__((ext_vector_type(2)));
typedef float v8f __attribute__((ext_vector_type(8)));

// D = A(16x4) * B(4x16) + C, all f32, wave32.
// 8-arg pattern (probe-confirmed): (neg_a, A, neg_b, B, c_mod, C, reuse_a, reuse_b)
__device__ __forceinline__ v8f wmma4(v2f a, v2f b, v8f c) {
    return __builtin_amdgcn_wmma_f32_16x16x4_f32(
        /*neg_a=*/false, a, /*neg_b=*/false, b,
        /*c_mod=*/(short)0, c, /*reuse_a=*/false, /*reuse_b=*/false);
}

// ---------------------------------------------------------------------------
// Weight layouts in workspace (all f32):
//  whp: stage-2 A operand, pre-swizzled.
//       whp[(((mt*32) + kq)*32 + lane)*2 + v] = Wh[mt*16 + (lane&15)][kq*4 + (lane>>4)*2 + v]
//       mt in {0,1}, kq in [0,32), lane in [0,32).  Size = 4096 floats.
//  wwp: stage-1 B operand, pre-swizzled (WwT padded to 32 cols with zeros).
//       wwp[q*64 + n*2 + j] = WwT[2*q + j][n] = Ww[n][2*q + j]   (0 for n >= 24)
//       q in [0,48), n in [0,32).  Size = 3072 floats.
// ---------------------------------------------------------------------------

// Triangle (bilinear) antialias weight, scale = 1/4 on both axes; matches
// jax.image.resize compute_weight_mat: sample_f = (i+0.5)*4 - 0.5,
// w = max(0, 1 - |j - sample_f| * 0.25), normalized over j.
__device__ __forceinline__ float tri_w(int i, int j) {
    float sf = 4.0f * (float)i + 1.5f;
    float d  = fabsf((float)j - sf) * 0.25f;
    return d < 1.0f ? (1.0f - d) : 0.0f;
}

__global__ __launch_bounds__(64) void init_weights_kernel(float* __restrict__ whp,
                                                          float* __restrict__ wwp) {
    int t = threadIdx.x;
    if (t < 32) {
        // Wh row i (output row), 128 input rows.
        int i = t;
        float sum = 0.0f;
        for (int j = 0; j < IN_H; ++j) sum += tri_w(i, j);
        float inv = 1.0f / sum;
        int mt = i >> 4;
        int m  = i & 15;
        for (int kq = 0; kq < 32; ++kq) {
            for (int kh = 0; kh < 2; ++kh) {
                int lane = kh * 16 + m;
                for (int v = 0; v < 2; ++v) {
                    int k = kq * 4 + kh * 2 + v;
                    whp[(((mt * 32) + kq) * 32 + lane) * 2 + v] = tri_w(i, k) * inv;
                }
            }
        }
    } else {
        // WwT column n (= output col of the W resize), 96 input cols; pad 24..31 with 0.
        int n = t - 32;                 // 0..31
        if (n < OUT_W) {
            float sum = 0.0f;
            for (int j = 0; j < IN_W; ++j) sum += tri_w(n, j);
            float inv = 1.0f / sum;
            for (int j = 0; j < IN_W; ++j)
                wwp[(j >> 1) * 64 + n * 2 + (j & 1)] = tri_w(n, j) * inv;
        } else {
            for (int j = 0; j < IN_W; ++j)
                wwp[(j >> 1) * 64 + n * 2 + (j & 1)] = 0.0f;
        }
    }
}

// ---------------------------------------------------------------------------
// Main kernel: one wave per image.  Y(32x24) = Wh(32x128) * X(128x96) * WwT(96x24)
// computed as 8 fused panels of 16 rows:
//   T_panel(16x32) = X[p*16:(p+1)*16, :] * WwT            (stage 1, 48 wmma)
//   Y             += Wh[:, p*16:(p+1)*16] * T_panel       (stage 2, 16 wmma)
// T_panel is bounced through LDS to convert the C/D lane layout into the
// B-operand lane layout (the only cross-lane shuffle needed).
// X is a strictly one-touch 603 MB stream -> non-temporal loads; output is
// write-once -> non-temporal stores.  Next panel is prefetched each iteration
// (global_prefetch_b8, no LOADcnt cost).
// ---------------------------------------------------------------------------
#define T_LD 36   // LDS row stride (dwords) for the 16x32 T panel

__global__ __launch_bounds__(WAVES_PER_BLOCK * 32) void resize_wmma_kernel(
        const float* __restrict__ x,
        const float* __restrict__ whp,
        const float* __restrict__ wwp,
        float* __restrict__ out,
        int nimg) {
    __shared__ float Tp[WAVES_PER_BLOCK][16 * T_LD];

    const int lane = threadIdx.x & 31;
    const int wave = threadIdx.x >> 5;
    int img = blockIdx.x * WAVES_PER_BLOCK + wave;
    if (img >= nimg) img = nimg - 1;   // uniform per wave; keeps barriers uniform

    const float* __restrict__ X = x + (size_t)img * IMG_ELEMS;
    float* __restrict__ tps = &Tp[wave][0];

    const int m  = lane & 15;   // M (A/C) or N (B) index
    const int kh = lane >> 4;   // K half-select

    v8f y00 = {}, y01 = {}, y10 = {}, y11 = {};

    for (int p = 0; p < 8; ++p) {
        // Prefetch next panel's row for this lane (covers the leading cachelines
        // of the next 6 KB tile; pure hint, no counter traffic).
        if (p < 7)
            __builtin_prefetch(X + ((p + 1) * 16 + m) * IN_W + kh * 2, 0, 1);

        // ---- stage 1: T_panel = X_panel(16x96) * WwT(96x32) ----
        v8f t0 = {}, t1 = {};
        const float* xrow = X + (p * 16 + m) * IN_W + kh * 2;  // this lane's A row base
        #pragma unroll
        for (int k = 0; k < 24; ++k) {                          // K-steps of 4 over 96
            v2f a  = __builtin_nontemporal_load((const v2f*)(xrow + k * 4));
            const float* wb = wwp + (k * 2 + kh) * 64;          // pre-swizzled B pairs
            v2f b0 = *(const v2f*)(wb + m * 2);                 // N-tile 0 (cols 0..15)
            v2f b1 = *(const v2f*)(wb + (16 + m) * 2);          // N-tile 1 (cols 16..31)
            t0 = wmma4(a, b0, t0);
            t1 = wmma4(a, b1, t1);
        }

        // ---- D layout -> LDS:  VGPR r holds (row = r + kh*8, col = m / 16+m) ----
        const int trow = kh * 8;
        #pragma unroll
        for (int r = 0; r < 8; ++r) {
            tps[(trow + r) * T_LD + m]      = t0[r];
            tps[(trow + r) * T_LD + 16 + m] = t1[r];
        }
        __syncthreads();

        // ---- stage 2: Y += Wh[:, p*16..] (32x16) * T_panel(16x32) ----
        #pragma unroll
        for (int kk = 0; kk < 4; ++kk) {                        // K-steps of 4 over 16
            const int kr = kk * 4 + kh * 2;
            v2f b0 = { tps[kr * T_LD + m],      tps[(kr + 1) * T_LD + m]      };
            v2f b1 = { tps[kr * T_LD + 16 + m], tps[(kr + 1) * T_LD + 16 + m] };
            const int kq = p * 4 + kk;                          // global K-quad over 128
            v2f a0 = *(const v2f*)(whp + ((0 * 32 + kq) * 32 + lane) * 2);
            v2f a1 = *(const v2f*)(whp + ((1 * 32 + kq) * 32 + lane) * 2);
            y00 = wmma4(a0, b0, y00);
            y01 = wmma4(a0, b1, y01);
            y10 = wmma4(a1, b0, y10);
            y11 = wmma4(a1, b1, y11);
        }
        __syncthreads();
    }

    // ---- store Y(32x24): C/D layout VGPR r -> (row = r + kh*8, col = lane&15) ----
    float* __restrict__ Y = out + (size_t)img * OUT_ELEMS;
    #pragma unroll
    for (int r = 0; r < 8; ++r) {
        const int row0 = r + kh * 8;
        __builtin_nontemporal_store(y00[r], Y + row0 * OUT_W + m);         // rows 0..15,  cols 0..15
        __builtin_nontemporal_store(y10[r], Y + (16 + row0) * OUT_W + m);  // rows 16..31, cols 0..15
        if (m < 8) {                                                        // cols 16..23
            __builtin_nontemporal_store(y01[r], Y + row0 * OUT_W + 16 + m);
            __builtin_nontemporal_store(y11[r], Y + (16 + row0) * OUT_W + 16 + m);
        }
    }
}

extern "C" void kernel_launch(void* const* d_in, const int* in_sizes, int n_in,
                              void* d_out, int out_size, void* d_ws, size_t ws_size,
                              hipStream_t stream) {
    const float* x = (const float*)d_in[0];
    float* out = (float*)d_out;

    // Workspace: whp (4096 f32) then wwp (3072 f32) = 28 KB.
    float* whp = (float*)d_ws;
    float* wwp = whp + 4096;

    const int nimg = in_sizes[0] / IMG_ELEMS;                 // 2048*6 = 12288
    const int blocks = (nimg + WAVES_PER_BLOCK - 1) / WAVES_PER_BLOCK;

    init_weights_kernel<<<1, 64, 0, stream>>>(whp, wwp);
    resize_wmma_kernel<<<blocks, WAVES_PER_BLOCK * 32, 0, stream>>>(x, whp, wwp, out, nimg);
}